// network_25658134626964
// MI455X (gfx1250) — compile-verified
//
#include <hip/hip_runtime.h>

typedef __attribute__((ext_vector_type(2))) float v2f;
typedef __attribute__((ext_vector_type(8))) float v8f;

#define HD 64

__device__ __forceinline__ float lrelu_f(float x) { return x > 0.f ? x : 0.1f * x; }

// Monotone (radix) float->uint encoding so atomicMax(u32) == float max.
__device__ __forceinline__ unsigned fenc(float f) {
    unsigned u = __float_as_uint(f);
    return (u & 0x80000000u) ? ~u : (u | 0x80000000u);
}
__device__ __forceinline__ float fdec(unsigned e) {
    unsigned u = (e & 0x80000000u) ? (e & 0x7fffffffu) : ~e;
    return __uint_as_float(u);
}

// ---------------------------------------------------------------------------
// Dense GEMM: out[rows x 64] = act( A[rows x 64] @ B[64 x 64]
//                                   (+ A2[rows x 16] @ B2[16 x 64])
//                                   (+ res) (+ bias) )
// One wave per 16-row tile. rows must be a multiple of 16 (100000, 4096 are).
// Uses V_WMMA_F32_16X16X4_F32, keeping full f32 precision.
// A layout  (16x4 f32): lanes L: M=L&15, a.x=A[M][k0+2*(L>>4)], a.y=next K.
// B layout  (4x16 f32): lanes L: N=L&15, b.x=B[k0+2*(L>>4)][N], b.y=next K.
// C/D layout: 8 VGPRs, lane L: N=L&15, VGPR v holds M = v + 8*(L>>4).
// ---------------------------------------------------------------------------
template <bool HAS_A2, bool HAS_RES, bool HAS_BIAS, int ACT> // ACT 0:none 1:lrelu 2:relu
__global__ __launch_bounds__(32) void gemm64(
    const float* __restrict__ A,  const float* __restrict__ B,
    const float* __restrict__ A2, const float* __restrict__ B2,
    const float* __restrict__ res, const float* __restrict__ bias,
    float* __restrict__ out)
{
    const int lane = threadIdx.x & 31;
    const int cn   = lane & 15;           // column within 16-wide tile / M row for A
    const int kh   = (lane >> 4) << 1;    // K sub-offset: 0 or 2
    const size_t row = (size_t)blockIdx.x * 16 + cn;

    const v8f z = {0.f, 0.f, 0.f, 0.f, 0.f, 0.f, 0.f, 0.f};
    v8f acc0 = z, acc1 = z, acc2 = z, acc3 = z;

    const float* arow = A + row * HD;
#pragma unroll
    for (int k0 = 0; k0 < 64; k0 += 4) {
        v2f a; a.x = arow[k0 + kh]; a.y = arow[k0 + kh + 1];
        const float* b0 = B + (size_t)(k0 + kh) * HD;
        const float* b1 = b0 + HD;
        v2f b;
        b.x = b0[cn];        b.y = b1[cn];
        acc0 = __builtin_amdgcn_wmma_f32_16x16x4_f32(false, a, false, b, (short)0, acc0, false, false);
        b.x = b0[16 + cn];   b.y = b1[16 + cn];
        acc1 = __builtin_amdgcn_wmma_f32_16x16x4_f32(false, a, false, b, (short)0, acc1, false, false);
        b.x = b0[32 + cn];   b.y = b1[32 + cn];
        acc2 = __builtin_amdgcn_wmma_f32_16x16x4_f32(false, a, false, b, (short)0, acc2, false, false);
        b.x = b0[48 + cn];   b.y = b1[48 + cn];
        acc3 = __builtin_amdgcn_wmma_f32_16x16x4_f32(false, a, false, b, (short)0, acc3, false, false);
    }
    if (HAS_A2) {
        const float* a2row = A2 + row * 16;
#pragma unroll
        for (int k0 = 0; k0 < 16; k0 += 4) {
            v2f a; a.x = a2row[k0 + kh]; a.y = a2row[k0 + kh + 1];
            const float* b0 = B2 + (size_t)(k0 + kh) * HD;
            const float* b1 = b0 + HD;
            v2f b;
            b.x = b0[cn];        b.y = b1[cn];
            acc0 = __builtin_amdgcn_wmma_f32_16x16x4_f32(false, a, false, b, (short)0, acc0, false, false);
            b.x = b0[16 + cn];   b.y = b1[16 + cn];
            acc1 = __builtin_amdgcn_wmma_f32_16x16x4_f32(false, a, false, b, (short)0, acc1, false, false);
            b.x = b0[32 + cn];   b.y = b1[32 + cn];
            acc2 = __builtin_amdgcn_wmma_f32_16x16x4_f32(false, a, false, b, (short)0, acc2, false, false);
            b.x = b0[48 + cn];   b.y = b1[48 + cn];
            acc3 = __builtin_amdgcn_wmma_f32_16x16x4_f32(false, a, false, b, (short)0, acc3, false, false);
        }
    }

    const size_t mbase = (size_t)blockIdx.x * 16 + ((lane >> 4) << 3);
    v8f accs[4] = {acc0, acc1, acc2, acc3};
#pragma unroll
    for (int t = 0; t < 4; ++t) {
        const int col = t * 16 + cn;
        const float bval = HAS_BIAS ? bias[col] : 0.f;
#pragma unroll
        for (int v = 0; v < 8; ++v) {
            const size_t m = mbase + v;
            float val = accs[t][v] + bval;
            if (HAS_RES) val += res[m * HD + col];
            if (ACT == 1) val = lrelu_f(val);
            else if (ACT == 2) val = fmaxf(val, 0.f);
            out[m * HD + col] = val;
        }
    }
}

// cE[l][d] = sum_k W_e[d][k] * a_e[l][k]   (4 x 16), 64 threads
__global__ void prep_ce(const float* __restrict__ W_e, const float* __restrict__ a_e,
                        float* __restrict__ cE)
{
    const int tid = threadIdx.x;              // tid = l*16 + d
    const int l = tid >> 4, d = tid & 15;
    float s = 0.f;
    for (int k = 0; k < 64; ++k) s += W_e[d * 64 + k] * a_e[l * 64 + k];
    cE[tid] = s;
}

// WeWm[l] = W_e @ Wm[l]  -> [4][16][64]
__global__ void prep_wewm(const float* __restrict__ W_e, const float* __restrict__ Wm,
                          float* __restrict__ WeWm)
{
    const int idx = blockIdx.x * blockDim.x + threadIdx.x;
    if (idx >= 4 * 16 * 64) return;
    const int l = idx >> 10, d = (idx >> 6) & 15, k = idx & 63;
    float s = 0.f;
    for (int j = 0; j < 64; ++j) s += W_e[d * 64 + j] * Wm[l * 4096 + j * 64 + k];
    WeWm[idx] = s;
}

// ceL[j][l] = r_edge[j] . cE[l]  for all 4 layers at once
__global__ void edge_ce(const float* __restrict__ re, const float* __restrict__ cE,
                        float* __restrict__ ceL, int E)
{
    const int j = blockIdx.x * blockDim.x + threadIdx.x;
    if (j >= E) return;
    float r[16];
    const float4* rp = (const float4*)(re + (size_t)j * 16);
#pragma unroll
    for (int i = 0; i < 4; ++i) {
        float4 v = rp[i];
        r[i * 4 + 0] = v.x; r[i * 4 + 1] = v.y; r[i * 4 + 2] = v.z; r[i * 4 + 3] = v.w;
    }
#pragma unroll
    for (int l = 0; l < 4; ++l) {
        float s = 0.f;
#pragma unroll
        for (int d = 0; d < 16; ++d) s += r[d] * cE[l * 16 + d];
        ceL[(size_t)j * 4 + l] = s;
    }
}

// hi = i_node * W_i (outer product), pooled = 0
__global__ void init_hi(const float* __restrict__ i_node, const float* __restrict__ W_i,
                        float* __restrict__ hi, float* __restrict__ pooled, int G)
{
    const int idx = blockIdx.x * blockDim.x + threadIdx.x;
    if (idx >= G * 64) return;
    hi[idx] = i_node[idx >> 6] * W_i[idx & 63];
    pooled[idx] = 0.f;
}

// Per-layer node prep: s = h.a_s, d = h.a_d; zero A64/A16/menc/denom rows.
__global__ void node_prep(const float* __restrict__ h, const float* __restrict__ as,
                          const float* __restrict__ ad, float* __restrict__ sb,
                          float* __restrict__ db, unsigned* __restrict__ menc,
                          float* __restrict__ denom, float* __restrict__ A64,
                          float* __restrict__ A16, int N)
{
    const int n = blockIdx.x * blockDim.x + threadIdx.x;
    if (n >= N) return;
    const float* hr = h + (size_t)n * 64;
    float s = 0.f, d = 0.f;
#pragma unroll
    for (int k = 0; k < 64; ++k) { const float v = hr[k]; s += v * as[k]; d += v * ad[k]; }
    sb[n] = s; db[n] = d; menc[n] = 0u; denom[n] = 0.f;
    const float4 z4 = {0.f, 0.f, 0.f, 0.f};
    float4* a64 = (float4*)(A64 + (size_t)n * 64);
#pragma unroll
    for (int i = 0; i < 16; ++i) a64[i] = z4;
    float4* a16 = (float4*)(A16 + (size_t)n * 16);
#pragma unroll
    for (int i = 0; i < 4; ++i) a16[i] = z4;
}

__global__ void edge_logits(const int* __restrict__ src, const int* __restrict__ dst,
                            const float* __restrict__ sb, const float* __restrict__ db,
                            const float* __restrict__ ceL, float* __restrict__ logit,
                            unsigned* __restrict__ menc, int E, int l)
{
    const int j = blockIdx.x * blockDim.x + threadIdx.x;
    if (j >= E) return;
    const float lg = lrelu_f(sb[src[j]] + db[dst[j]] + ceL[(size_t)j * 4 + l]);
    logit[j] = lg;
    atomicMax(menc + dst[j], fenc(lg));
}

__global__ void edge_exp(const int* __restrict__ dst, const unsigned* __restrict__ menc,
                         float* __restrict__ logit, float* __restrict__ denom, int E)
{
    const int j = blockIdx.x * blockDim.x + threadIdx.x;
    if (j >= E) return;
    const int d = dst[j];
    const float ex = __expf(logit[j] - fdec(menc[d]));
    logit[j] = ex;
    atomicAdd(denom + d, ex);
}

// 64 threads per edge: A64[dst] += attn*h[src]; A16[dst] += attn*r_edge[j]
__global__ __launch_bounds__(256) void edge_scatter(
    const int* __restrict__ src, const int* __restrict__ dst,
    const float* __restrict__ logit, const float* __restrict__ denom,
    const float* __restrict__ h, const float* __restrict__ re,
    float* __restrict__ A64, float* __restrict__ A16, int E)
{
    const int j = blockIdx.x * 4 + (threadIdx.x >> 6);
    const int t = threadIdx.x & 63;
    if (j >= E) return;
    const int s = src[j], d = dst[j];
    const float attn = logit[j] / denom[d];
    atomicAdd(A64 + (size_t)d * 64 + t, attn * h[(size_t)s * 64 + t]);
    if (t < 16) atomicAdd(A16 + (size_t)d * 16 + t, attn * re[(size_t)j * 16 + t]);
}

__global__ void gate_hi(const float* __restrict__ h, const float* __restrict__ d_edge,
                        const int* __restrict__ gid, const float* __restrict__ wd,
                        const float* __restrict__ bd, float* __restrict__ gate,
                        float* __restrict__ hi, int N)
{
    const int idx = blockIdx.x * blockDim.x + threadIdx.x;
    if (idx >= N * 64) return;
    const int n = idx >> 6, k = idx & 63;
    const float g = 1.f / (1.f + __expf(-(d_edge[n] * wd[0] + bd[0])));
    if (k == 0) gate[n] = g;
    atomicAdd(hi + (size_t)gid[n] * 64 + k, g * h[idx]);
}

__global__ void h_mix_pool(float* __restrict__ h, const float* __restrict__ gate,
                           const float* __restrict__ hi, const int* __restrict__ gid,
                           float* __restrict__ pooled, int N)
{
    const int idx = blockIdx.x * blockDim.x + threadIdx.x;
    if (idx >= N * 64) return;
    const int n = idx >> 6, k = idx & 63;
    const float v = h[idx] + gate[n] * hi[(size_t)gid[n] * 64 + k];
    h[idx] = v;
    atomicAdd(pooled + (size_t)gid[n] * 64 + k, v);
}

__global__ void out_kernel(const float* __restrict__ x, const float* __restrict__ Wo,
                           const float* __restrict__ bo, float* __restrict__ out, int G)
{
    const int g = blockIdx.x * blockDim.x + threadIdx.x;
    if (g >= G) return;
    float s = bo[0];
#pragma unroll
    for (int k = 0; k < 64; ++k) s += x[(size_t)g * 64 + k] * Wo[k];
    out[g] = s;
}

extern "C" void kernel_launch(void* const* d_in, const int* in_sizes, int n_in,
                              void* d_out, int out_size, void* d_ws, size_t ws_size,
                              hipStream_t stream)
{
    const float* r_node = (const float*)d_in[0];
    const float* i_node = (const float*)d_in[1];
    const float* r_edge = (const float*)d_in[2];
    const float* d_edge = (const float*)d_in[3];
    const int*   src    = (const int*)d_in[4];
    const int*   dst    = (const int*)d_in[5];
    const int*   gid    = (const int*)d_in[6];
    const float* W_r    = (const float*)d_in[7];
    const float* W_i    = (const float*)d_in[8];
    const float* W_e    = (const float*)d_in[9];
    const float* Wm     = (const float*)d_in[10];
    const float* a_s    = (const float*)d_in[11];
    const float* a_d    = (const float*)d_in[12];
    const float* a_e    = (const float*)d_in[13];
    const float* w_d    = (const float*)d_in[14];
    const float* b_d    = (const float*)d_in[15];
    const float* W_mlp  = (const float*)d_in[16];
    const float* b_mlp  = (const float*)d_in[17];
    const float* W_out  = (const float*)d_in[18];
    const float* b_out  = (const float*)d_in[19];

    const int N = in_sizes[0] / 64;   // 100000 (multiple of 16)
    const int G = in_sizes[1];        // 4096
    const int E = in_sizes[2] / 16;   // 1200000
    (void)n_in; (void)out_size; (void)ws_size;

    float* ws = (float*)d_ws;
    size_t off = 0;
    auto carve = [&](size_t n) -> float* {
        float* p = ws + off; off += (n + 63) & ~(size_t)63; return p;
    };
    float*    h      = carve((size_t)N * 64);
    float*    A64    = carve((size_t)N * 64);
    float*    A16    = carve((size_t)N * 16);
    float*    sb     = carve(N);
    float*    db     = carve(N);
    unsigned* menc   = (unsigned*)carve(N);
    float*    denom  = carve(N);
    float*    gate   = carve(N);
    float*    ceL    = carve((size_t)E * 4);
    float*    logit  = carve(E);
    float*    hi     = carve((size_t)G * 64);
    float*    pooled = carve((size_t)G * 64);
    float*    x1     = carve((size_t)G * 64);
    float*    x2     = carve((size_t)G * 64);
    float*    cE     = carve(64);
    float*    WeWm   = carve(4 * 16 * 64);

    // One-time small precomputes (cheap, every call for determinism)
    hipLaunchKernelGGL(prep_ce, dim3(1), dim3(64), 0, stream, W_e, a_e, cE);
    hipLaunchKernelGGL(prep_wewm, dim3((4096 + 255) / 256), dim3(256), 0, stream, W_e, Wm, WeWm);
    hipLaunchKernelGGL(edge_ce, dim3((E + 255) / 256), dim3(256), 0, stream, r_edge, cE, ceL, E);

    // h = r_node @ W_r   (WMMA)
    hipLaunchKernelGGL((gemm64<false, false, false, 0>), dim3(N / 16), dim3(32), 0, stream,
                       r_node, W_r, nullptr, nullptr, nullptr, nullptr, h);
    hipLaunchKernelGGL(init_hi, dim3((G * 64 + 255) / 256), dim3(256), 0, stream,
                       i_node, W_i, hi, pooled, G);

    for (int l = 0; l < 4; ++l) {
        hipLaunchKernelGGL(node_prep, dim3((N + 127) / 128), dim3(128), 0, stream,
                           h, a_s + l * 64, a_d + l * 64, sb, db, menc, denom, A64, A16, N);
        hipLaunchKernelGGL(edge_logits, dim3((E + 255) / 256), dim3(256), 0, stream,
                           src, dst, sb, db, ceL, logit, menc, E, l);
        hipLaunchKernelGGL(edge_exp, dim3((E + 255) / 256), dim3(256), 0, stream,
                           dst, menc, logit, denom, E);
        hipLaunchKernelGGL(edge_scatter, dim3((E + 3) / 4), dim3(256), 0, stream,
                           src, dst, logit, denom, h, r_edge, A64, A16, E);
        // h = lrelu(h + A64@Wm[l] + A16@(W_e@Wm[l]))   (WMMA)
        hipLaunchKernelGGL((gemm64<true, true, false, 1>), dim3(N / 16), dim3(32), 0, stream,
                           A64, Wm + (size_t)l * 4096, A16, WeWm + (size_t)l * 1024,
                           h, nullptr, h);
    }

    hipLaunchKernelGGL(gate_hi, dim3((N * 64 + 255) / 256), dim3(256), 0, stream,
                       h, d_edge, gid, w_d, b_d, gate, hi, N);
    hipLaunchKernelGGL(h_mix_pool, dim3((N * 64 + 255) / 256), dim3(256), 0, stream,
                       h, gate, hi, gid, pooled, N);

    // MLP readout (WMMA, bias + relu)
    hipLaunchKernelGGL((gemm64<false, false, true, 2>), dim3(G / 16), dim3(32), 0, stream,
                       pooled, W_mlp + 0 * 4096, nullptr, nullptr, nullptr, b_mlp + 0 * 64, x1);
    hipLaunchKernelGGL((gemm64<false, false, true, 2>), dim3(G / 16), dim3(32), 0, stream,
                       x1, W_mlp + 1 * 4096, nullptr, nullptr, nullptr, b_mlp + 1 * 64, x2);
    hipLaunchKernelGGL((gemm64<false, false, true, 2>), dim3(G / 16), dim3(32), 0, stream,
                       x2, W_mlp + 2 * 4096, nullptr, nullptr, nullptr, b_mlp + 2 * 64, x1);

    hipLaunchKernelGGL(out_kernel, dim3((G + 255) / 256), dim3(256), 0, stream,
                       x1, W_out, b_out, (float*)d_out, G);
}